// DropBlock2D_18133351924169
// MI455X (gfx1250) — compile-verified
//
#include <hip/hip_runtime.h>
#include <stdint.h>

// ---------------- problem constants ----------------
#define HH 56
#define WW 56
#define NIMG (64 * 256)                 // B*C = 16384 independent 56x56 images
#define IMG_ELEMS (HH * WW)             // 3136
#define TOTAL_ELEMS (NIMG * IMG_ELEMS)  // 51,380,224
#define MASK56 ((1ull << 56) - 1ull)

// gamma = (1-0.9)/49 * (56/50)^2 = 0.00256 exactly
#define GAMMA 0.00256f

typedef float v4f __attribute__((ext_vector_type(4)));

// ============================================================
// Pass 1: one workgroup per (b,c) image.
//  - async-DMA u tile (12544 B) into LDS (non-temporal: u is read once,
//    keep L2 free for the drop-bit array consumed by pass 2)
//  - ballot -> 56-bit row masks of Bernoulli hits
//  - 7-wide horizontal OR (bit shifts) + 7-tall vertical OR
//  - emit 56 uint64 "drop" words per image + exact integer keep-count
// ============================================================
__global__ __launch_bounds__(256) void dropblock_mask_kernel(
    const float* __restrict__ u,
    unsigned long long* __restrict__ bitsOut,
    unsigned int* __restrict__ counter)
{
    __shared__ __align__(16) float su[IMG_ELEMS];     // staged u tile
    __shared__ unsigned int rowbits[112];             // 56 rows x 2 half-words
    __shared__ unsigned long long hbuf[HH];           // horizontally dilated rows
    __shared__ unsigned int sh_cnt;

    const int img  = blockIdx.x;
    const int tid  = threadIdx.x;
    const int wave = tid >> 5;
    const int lane = tid & 31;

    if (tid == 0) sh_cnt = 0u;

    // ---- async stage: global u -> LDS, 16 B per lane per op, NT hint ----
    const float* ubase = u + (size_t)img * IMG_ELEMS;
    for (int t = tid; t < IMG_ELEMS / 4; t += 256) {
        const float* gaddr = ubase + t * 4;
        unsigned int loff = (unsigned int)(uintptr_t)(&su[t * 4]); // low 32b of generic ptr == LDS addr
        asm volatile("global_load_async_to_lds_b128 %0, %1, off th:TH_LOAD_NT"
                     :: "v"(loff), "v"(gaddr) : "memory");
    }
    asm volatile("s_wait_asynccnt 0" ::: "memory");
    __syncthreads();

    // ---- ballot phase: 112 32-lane column chunks, 14 per wave ----
    for (int it = 0; it < 14; ++it) {
        int ch  = wave * 14 + it;
        int row = ch >> 1;
        int col = ((ch & 1) << 5) + lane;
        int cc  = col < WW ? col : WW - 1;           // clamp (select, no divergence)
        float v = su[row * WW + cc];
        bool hit = (col < WW) && (v < GAMMA);
        unsigned m = __builtin_amdgcn_ballot_w32(hit);
        if (lane == 0) rowbits[ch] = m;
    }
    __syncthreads();

    // ---- horizontal dilation: OR of 7 shifted copies ----
    if (tid < HH) {
        unsigned long long m = (unsigned long long)rowbits[2 * tid] |
                               ((unsigned long long)rowbits[2 * tid + 1] << 32);
        m &= MASK56;
        unsigned long long h = m | (m << 1) | (m << 2) | (m << 3)
                                 | (m >> 1) | (m >> 2) | (m >> 3);
        hbuf[tid] = h;
    }
    __syncthreads();

    // ---- vertical dilation + emit drop bits + count kept pixels ----
    if (tid < HH) {
        int lo = tid - 3 < 0 ? 0 : tid - 3;
        int hi = tid + 3 > HH - 1 ? HH - 1 : tid + 3;
        unsigned long long v = 0ull;
        for (int r = lo; r <= hi; ++r) v |= hbuf[r];
        v &= MASK56;
        bitsOut[(size_t)img * HH + tid] = v;          // bit set == dropped (RT -> stays in L2)
        unsigned zeros = (unsigned)(WW - __popcll(v)); // kept pixels in this row
        atomicAdd(&sh_cnt, zeros);
    }
    __syncthreads();
    if (tid == 0) atomicAdd(counter, sh_cnt);
}

// ============================================================
// Pass 2: out = x * (dropped ? 0 : N/count), float4-vectorized,
// non-temporal on the 410 MB x/out streams so the 7.3 MB bit
// array + counter stay hot in the 192 MB L2.
// ============================================================
__global__ __launch_bounds__(256) void dropblock_apply_kernel(
    const float* __restrict__ x,
    const unsigned long long* __restrict__ bits,
    const unsigned int* __restrict__ counter,
    float* __restrict__ out)
{
    const unsigned idx = blockIdx.x * 256u + threadIdx.x;   // float4 index
    if (idx >= (unsigned)(TOTAL_ELEMS / 4)) return;

    const unsigned img = idx / 784u;            // 784 float4 per image
    const unsigned rem = idx - img * 784u;
    const unsigned r   = rem / 14u;             // 14 float4 per row
    const unsigned c4  = rem - r * 14u;

    const float scale = (float)TOTAL_ELEMS / (float)(*counter); // uniform -> scalar load

    unsigned long long b = bits[(size_t)img * HH + r] >> (c4 * 4u);

    const v4f* x4 = (const v4f*)x;
    v4f xv = __builtin_nontemporal_load(&x4[idx]);
    v4f ov;
#pragma unroll
    for (int j = 0; j < 4; ++j)
        ov[j] = ((b >> j) & 1ull) ? 0.0f : xv[j] * scale;

    v4f* o4 = (v4f*)out;
    __builtin_nontemporal_store(ov, &o4[idx]);
}

// ============================================================
extern "C" void kernel_launch(void* const* d_in, const int* in_sizes, int n_in,
                              void* d_out, int out_size, void* d_ws, size_t ws_size,
                              hipStream_t stream)
{
    const float* x = (const float*)d_in[0];
    const float* u = (const float*)d_in[1];
    float* out = (float*)d_out;

    // workspace layout: [0..63] counter (zeroed each call), [64..] bit array
    unsigned int* counter = (unsigned int*)d_ws;
    unsigned long long* bits = (unsigned long long*)((char*)d_ws + 64);

    hipMemsetAsync(d_ws, 0, 64, stream);

    dropblock_mask_kernel<<<NIMG, 256, 0, stream>>>(u, bits, counter);

    const int n4 = TOTAL_ELEMS / 4;             // 12,845,056
    dropblock_apply_kernel<<<n4 / 256, 256, 0, stream>>>(x, bits, counter, out);
}